// BiaffineLayerv1_11201274708613
// MI455X (gfx1250) — compile-verified
//
#include <hip/hip_runtime.h>
#include <hip/hip_bf16.h>

typedef __attribute__((ext_vector_type(2))) float v2f;
typedef __attribute__((ext_vector_type(4))) float v4f;
typedef __attribute__((ext_vector_type(8))) float v8f;

#define BDIM 2
#define LDIM 512
#define HDIM 128
#define CDIM 25

// K axis (512): [0,128)=t3, [128,256)=t4, [256,384)=t1, [384,512)=t2.  Bias added in epilogue.
#define BL_STRIDE 520       // B-panel LDS row stride (floats), even + conflict-friendly, 16B-aligned rows
#define X2_STRIDE 132       // x2 tile LDS row stride (floats), 16B-aligned rows

__global__ __launch_bounds__(512) void biaffine_wmma_f32(
    const float* __restrict__ x1, const float* __restrict__ x2,
    const float* __restrict__ W,  const float* __restrict__ bias,
    float* __restrict__ out)
{
    __shared__ float Bl [32 * BL_STRIDE];   // Wcat[c=0..31][k=0..511] (rows c>=25 zero)
    __shared__ float x1t[16 * HDIM];        // x1[l0..l0+15][h]
    __shared__ float x2t[16 * X2_STRIDE];   // x2[m0..m0+15][h] (padded stride)

    const int tid = threadIdx.x;
    const int mt  = blockIdx.x;             // m tile: 0..31
    const int lt  = blockIdx.y;             // l tile: 0..31
    const int b   = blockIdx.z;             // batch
    const int l0  = lt * 16, m0 = mt * 16;

    // ---- stage B panel (float4): Wcat[c][k] = W[c][(k+256)&511] = W3,W4,W1,W2 concat ----
    // 32 rows x 128 float4 = 4096 quads; 512 threads -> 8 iterations; no integer division.
#pragma unroll
    for (int it = 0; it < 8; ++it) {
        const int idx = tid + it * 512;
        const int c   = idx >> 7;            // 0..31
        const int k   = (idx & 127) << 2;    // 0..508, step 4 (16B groups never straddle the rotate)
        v4f v = {};
        if (c < CDIM)
            v = *(const v4f*)&W[c * 512 + ((k + 256) & 511)];
        *(v4f*)&Bl[c * BL_STRIDE + k] = v;
    }
    // ---- stage x1 / x2 tiles (one float4 per thread per tile) ----
    {
        const int r  = tid >> 5;             // 0..15
        const int h4 = (tid & 31) << 2;      // 0..124, step 4
        *(v4f*)&x1t[r * HDIM      + h4] = *(const v4f*)&x1[((size_t)b * LDIM + l0 + r) * HDIM + h4];
        *(v4f*)&x2t[r * X2_STRIDE + h4] = *(const v4f*)&x2[((size_t)b * LDIM + m0 + r) * HDIM + h4];
    }
    __syncthreads();

    const int wave = tid >> 5;      // 0..15 -> fixed l row for this wave
    const int lane = tid & 31;
    const int g    = lane >> 4;     // lane half: K pair {2g, 2g+1} within each 4-wide step
    const int n    = lane & 15;     // A: M (m row index); B: N (c column)

    // Lane-local base pointers with the 2*g K-offset folded in; loop stride is 4 floats.
    const float* __restrict__ x1p = &x1t[wave * HDIM          + 2 * g];
    const float* __restrict__ x2p = &x2t[n * X2_STRIDE        + 2 * g];
    const float* __restrict__ b0p = &Bl [n * BL_STRIDE        + 2 * g];
    const float* __restrict__ b1p = &Bl [(n + 16) * BL_STRIDE + 2 * g];

    v8f acc0 = {};  // c = 0..15
    v8f acc1 = {};  // c = 16..31 (only 16..24 stored)

#define WMMA_STEP(AX, AY, KOFF)                                                        \
    do {                                                                               \
        v2f a;  a.x = (AX); a.y = (AY);                                                \
        v2f bf0; bf0.x = b0p[(KOFF)]; bf0.y = b0p[(KOFF) + 1];                         \
        v2f bf1; bf1.x = b1p[(KOFF)]; bf1.y = b1p[(KOFF) + 1];                         \
        acc0 = __builtin_amdgcn_wmma_f32_16x16x4_f32(false, a, false, bf0, (short)0,   \
                                                     acc0, false, false);              \
        acc1 = __builtin_amdgcn_wmma_f32_16x16x4_f32(false, a, false, bf1, (short)0,   \
                                                     acc1, false, false);              \
    } while (0)

    // Segment 1: k in [0,128)   A = x1 * x2            (t3 / W3)
#pragma unroll 4
    for (int s = 0; s < 32; ++s) {
        const int k = 4 * s;
        WMMA_STEP(x1p[k] * x2p[k], x1p[k + 1] * x2p[k + 1], k);
    }
    // Segment 2: k in [128,256) A = |x1 - x2|          (t4 / W4)
#pragma unroll 4
    for (int s = 0; s < 32; ++s) {
        const int k = 4 * s;
        WMMA_STEP(fabsf(x1p[k] - x2p[k]), fabsf(x1p[k + 1] - x2p[k + 1]), 128 + k);
    }
    // Segment 3: k in [256,384) A = x1 (broadcast over m)  (t1 / W1)
#pragma unroll 4
    for (int s = 0; s < 32; ++s) {
        const int k = 4 * s;
        WMMA_STEP(x1p[k], x1p[k + 1], 256 + k);
    }
    // Segment 4: k in [384,512) A = x2                  (t2 / W2)
#pragma unroll 4
    for (int s = 0; s < 32; ++s) {
        const int k = 4 * s;
        WMMA_STEP(x2p[k], x2p[k + 1], 384 + k);
    }
#undef WMMA_STEP

    // ---- epilogue: D rows + bias -> out[b][l][m][c] ----
    const float bias0 = bias[n];
    const float bias1 = (n < CDIM - 16) ? bias[16 + n] : 0.0f;

    const int l = l0 + wave;
    float* __restrict__ outbase = out + ((size_t)b * LDIM + l) * (size_t)LDIM * CDIM;
#pragma unroll
    for (int v = 0; v < 8; ++v) {
        const int m = m0 + v + 8 * g;            // D VGPR v: lanes 0-15 -> M=v, lanes 16-31 -> M=v+8
        const size_t o = (size_t)m * CDIM;
        outbase[o + n] = acc0[v] + bias0;        // c = n
        if (n < CDIM - 16)                       // c = 16+n, valid for n < 9
            outbase[o + 16 + n] = acc1[v] + bias1;
    }
}

extern "C" void kernel_launch(void* const* d_in, const int* in_sizes, int n_in,
                              void* d_out, int out_size, void* d_ws, size_t ws_size,
                              hipStream_t stream) {
    (void)in_sizes; (void)n_in; (void)d_ws; (void)ws_size; (void)out_size;
    const float* x1   = (const float*)d_in[0];
    const float* x2   = (const float*)d_in[1];
    const float* W    = (const float*)d_in[2];
    const float* bias = (const float*)d_in[3];
    float* out = (float*)d_out;

    dim3 grid(LDIM / 16, LDIM / 16, BDIM);   // (m tiles, l tiles, batch) = (32, 32, 2)
    dim3 block(512);                          // 16 wave32s
    biaffine_wmma_f32<<<grid, block, 0, stream>>>(x1, x2, W, bias, out);
}